// PointNetBackbone_32461362823174
// MI455X (gfx1250) — compile-verified
//
#include <hip/hip_runtime.h>

typedef __attribute__((ext_vector_type(2))) float v2f;
typedef __attribute__((ext_vector_type(8))) float v8f;

__device__ __forceinline__ v8f wmma_f32_16x16x4(v2f a, v2f b, v8f c) {
  // D = A(16x4 f32) x B(4x16 f32) + C(16x16 f32), full fp32 precision.
  return __builtin_amdgcn_wmma_f32_16x16x4_f32(false, a, false, b, (short)0, c,
                                               false, false);
}

// ---------------------------------------------------------------------------
// Pass-through copies: coords[0] = input[:, :, :3], data[0] = input
// ---------------------------------------------------------------------------
__global__ __launch_bounds__(256) void copy_strided3_kernel(
    const float* __restrict__ in, float* __restrict__ out, int rows) {
  int i = blockIdx.x * 256 + threadIdx.x;
  if (i < rows * 3) {
    int r = i / 3, c = i - r * 3;
    out[i] = in[r * 9 + c];
  }
}

__global__ __launch_bounds__(256) void copy_flat_kernel(
    const float* __restrict__ in, float* __restrict__ out, long long n) {
  long long i = (long long)blockIdx.x * 256 + threadIdx.x;
  if (i < n) out[i] = in[i];
}

// ---------------------------------------------------------------------------
// Farthest point sampling. One workgroup per batch. Running min-distance
// array lives in (dynamic) LDS; argmax per iteration = wave32 shfl_xor
// reduction + 8-partial cross-wave reduce. Also emits new_xyz directly.
// ---------------------------------------------------------------------------
__global__ __launch_bounds__(256) void fps_kernel(
    const float* __restrict__ coords, int cstride, int N, int S,
    int* __restrict__ idxout, float* __restrict__ newxyz) {
  extern __shared__ float dist[];  // N floats
  __shared__ float pd[8];
  __shared__ int pi[8];
  __shared__ float cen[3];
  __shared__ int sFar;

  const int b = blockIdx.x;
  const int tid = threadIdx.x;
  const float* base = coords + (long long)b * N * cstride;

  for (int i = tid; i < N; i += 256) dist[i] = 1e10f;
  if (tid == 0) sFar = 0;
  __syncthreads();

  for (int it = 0; it < S; ++it) {
    if (tid == 0) {
      int far = sFar;
      idxout[b * S + it] = far;
      float cx = base[(long long)far * cstride + 0];
      float cy = base[(long long)far * cstride + 1];
      float cz = base[(long long)far * cstride + 2];
      cen[0] = cx; cen[1] = cy; cen[2] = cz;
      newxyz[(long long)(b * S + it) * 3 + 0] = cx;
      newxyz[(long long)(b * S + it) * 3 + 1] = cy;
      newxyz[(long long)(b * S + it) * 3 + 2] = cz;
    }
    __syncthreads();
    const float cx = cen[0], cy = cen[1], cz = cen[2];

    float bd = -1.0f;
    int bi = 0;
    for (int i = tid; i < N; i += 256) {
      float dx = base[(long long)i * cstride + 0] - cx;
      float dy = base[(long long)i * cstride + 1] - cy;
      float dz = base[(long long)i * cstride + 2] - cz;
      float d = dx * dx + dy * dy + dz * dz;
      float nd = fminf(dist[i], d);
      dist[i] = nd;
      if (nd > bd) { bd = nd; bi = i; }  // '>' keeps first index on ties
    }
    // wave32 argmax reduction (first index wins on ties)
    #pragma unroll
    for (int o = 16; o > 0; o >>= 1) {
      float od = __shfl_xor(bd, o, 32);
      int oi = __shfl_xor(bi, o, 32);
      if (od > bd || (od == bd && oi < bi)) { bd = od; bi = oi; }
    }
    if ((tid & 31) == 0) { pd[tid >> 5] = bd; pi[tid >> 5] = bi; }
    __syncthreads();
    if (tid == 0) {
      float m = pd[0]; int mi = pi[0];
      for (int w = 1; w < 8; ++w)
        if (pd[w] > m || (pd[w] == m && pi[w] < mi)) { m = pd[w]; mi = pi[w]; }
      sFar = mi;
    }
    __syncthreads();
  }
}

// ---------------------------------------------------------------------------
// Ball query: one wave per centroid. Scan points in index order in chunks of
// 32 using ballot + prefix popcount; pad with first hit (matches sort+pad).
// Distance uses the reference's ||q||^2+||p||^2-2q.p expansion.
// ---------------------------------------------------------------------------
__global__ __launch_bounds__(256) void ballquery_kernel(
    const float* __restrict__ coords, int cstride, int N, int S,
    const float* __restrict__ newxyz, float r2, int* __restrict__ outidx,
    int total) {
  const int g = blockIdx.x * 8 + (threadIdx.x >> 5);
  const int lane = threadIdx.x & 31;
  if (g >= total) return;
  const int b = g / S;
  const float* base = coords + (long long)b * N * cstride;
  const float qx = newxyz[g * 3 + 0];
  const float qy = newxyz[g * 3 + 1];
  const float qz = newxyz[g * 3 + 2];
  const float qq = qx * qx + qy * qy + qz * qz;

  int cnt = 0;
  int firstIdx = -1;
  for (int bs = 0; bs < N && cnt < 32; bs += 32) {
    const int p = bs + lane;  // N is always a multiple of 32
    float px = base[(long long)p * cstride + 0];
    float py = base[(long long)p * cstride + 1];
    float pz = base[(long long)p * cstride + 2];
    float sq = qq + (px * px + py * py + pz * pz) -
               2.0f * (qx * px + qy * py + qz * pz);
    bool in = !(sq > r2);
    unsigned long long m = __ballot(in);
    int pre = __popcll(m & ((1ull << lane) - 1ull));
    if (in && (cnt + pre) < 32) outidx[g * 32 + cnt + pre] = p;
    if (in && (cnt + pre) == 0) firstIdx = p;
    cnt += __popcll(m);
  }
  #pragma unroll
  for (int o = 16; o > 0; o >>= 1)
    firstIdx = max(firstIdx, __shfl_xor(firstIdx, o, 32));
  if (cnt > 32) cnt = 32;
  if (lane >= cnt) outidx[g * 32 + lane] = firstIdx;
}

// ---------------------------------------------------------------------------
// Fused group-gather + 3-layer MLP (fp32 WMMA) + max-pool.
// One workgroup (8 waves) per group (b,s); M = 32 samples = 2 M-tiles.
// Layer1 A fragments gathered from global via ball indices; activations for
// layers 2/3 staged in LDS with conflict-free (+4 % 64) row padding.
// Layer 3 fuses ReLU + max-pool in registers.
// ---------------------------------------------------------------------------
template <int C0, int C0P, int C1, int C2, int C3>
__global__ __launch_bounds__(256) void sa_mlp_kernel(
    const float* __restrict__ coords, int cstride,
    const float* __restrict__ feats, int fstride,
    const float* __restrict__ newxyz, const int* __restrict__ ballidx,
    const float* __restrict__ W1, const float* __restrict__ B1,
    const float* __restrict__ W2, const float* __restrict__ B2,
    const float* __restrict__ W3, const float* __restrict__ B3,
    float* __restrict__ outfeat, int N, int S) {
  constexpr int PAD = ((32 * (C1 + 4) + 32 * (C2 + 4)) * 4 <= 65536) ? 4 : 0;
  constexpr int LDA1 = C1 + PAD;
  constexpr int LDA2 = C2 + PAD;
  __shared__ float sA[32 * LDA1];
  __shared__ float sB[32 * LDA2];

  const int g = blockIdx.x;
  const int tid = threadIdx.x;
  const int lane = tid & 31;
  const int wave = tid >> 5;
  const int ml = lane & 15;            // M (or N) index inside a 16-tile
  const int khalf = (lane >> 4) << 1;  // K offset: lanes 0-15 -> 0, 16-31 -> 2
  const int rhalf = (lane >> 4) << 3;  // D-tile row offset: 0 or 8

  // idx + centroid carved from sB (sB is untouched until layer 2)
  int* idx_s = (int*)sB;
  float* nx = sB + 32;
  if (tid < 32) idx_s[tid] = ballidx[g * 32 + tid];
  if (tid < 3) nx[tid] = newxyz[g * 3 + tid];
  __syncthreads();

  const int bN = (g / S) * N;

  // ---- Layer 1: gathered X (32 x C0P, zero-padded K) @ W1 -> sA ------------
  {
    const int tiles = 2 * (C1 / 16);
    for (int t = wave; t < tiles; t += 8) {
      const int mt = t & 1, nt = t >> 1;
      const int m = mt * 16 + ml;
      const int n = nt * 16 + ml;
      const int p = idx_s[m];
      const long long rowc = (long long)(bN + p) * cstride;
      const long long rowf = (long long)(bN + p) * fstride;
      auto xval = [&](int kk) -> float {
        if (kk < 3) return coords[rowc + kk] - nx[kk];
        if (kk < C0) return feats[rowf + (kk - 3)];
        return 0.0f;
      };
      v8f acc = {};
      for (int k0 = 0; k0 < C0P; k0 += 4) {
        const int kk = k0 + khalf;
        v2f a, w;
        a.x = xval(kk);
        a.y = xval(kk + 1);
        w.x = (kk < C0) ? W1[kk * C1 + n] : 0.0f;
        w.y = (kk + 1 < C0) ? W1[(kk + 1) * C1 + n] : 0.0f;
        acc = wmma_f32_16x16x4(a, w, acc);
      }
      const float bias = B1[n];
      #pragma unroll
      for (int j = 0; j < 8; ++j) {
        const int row = mt * 16 + j + rhalf;
        sA[row * LDA1 + n] = fmaxf(acc[j] + bias, 0.0f);
      }
    }
  }
  __syncthreads();

  // ---- Layer 2: sA (32 x C1) @ W2 -> sB -----------------------------------
  {
    const int tiles = 2 * (C2 / 16);
    for (int t = wave; t < tiles; t += 8) {
      const int mt = t & 1, nt = t >> 1;
      const int m = mt * 16 + ml;
      const int n = nt * 16 + ml;
      v8f acc = {};
      for (int k0 = 0; k0 < C1; k0 += 4) {
        const int kk = k0 + khalf;
        v2f a = {sA[m * LDA1 + kk], sA[m * LDA1 + kk + 1]};
        v2f w = {W2[kk * C2 + n], W2[(kk + 1) * C2 + n]};
        acc = wmma_f32_16x16x4(a, w, acc);
      }
      const float bias = B2[n];
      #pragma unroll
      for (int j = 0; j < 8; ++j) {
        const int row = mt * 16 + j + rhalf;
        sB[row * LDA2 + n] = fmaxf(acc[j] + bias, 0.0f);
      }
    }
  }
  __syncthreads();

  // ---- Layer 3: sB (32 x C2) @ W3, fused ReLU + max over 32 samples -------
  float* sMax = sA;  // reuse (2 * C3 floats <= 32 * LDA1)
  {
    const int tiles = 2 * (C3 / 16);
    for (int t = wave; t < tiles; t += 8) {
      const int mt = t & 1, nt = t >> 1;
      const int m = mt * 16 + ml;
      const int n = nt * 16 + ml;
      v8f acc = {};
      for (int k0 = 0; k0 < C2; k0 += 4) {
        const int kk = k0 + khalf;
        v2f a = {sB[m * LDA2 + kk], sB[m * LDA2 + kk + 1]};
        v2f w = {W3[kk * C3 + n], W3[(kk + 1) * C3 + n]};
        acc = wmma_f32_16x16x4(a, w, acc);
      }
      const float bias = B3[n];
      float pm = 0.0f;  // ReLU floor doubles as max identity
      #pragma unroll
      for (int j = 0; j < 8; ++j) pm = fmaxf(pm, acc[j] + bias);
      pm = fmaxf(pm, __shfl_xor(pm, 16, 32));  // combine rows j and j+8
      if (lane < 16) sMax[mt * C3 + n] = pm;
    }
  }
  __syncthreads();

  for (int c = tid; c < C3; c += 256)
    outfeat[(long long)g * C3 + c] = fmaxf(sMax[c], sMax[C3 + c]);
}

// ---------------------------------------------------------------------------
extern "C" void kernel_launch(void* const* d_in, const int* in_sizes, int n_in,
                              void* d_out, int out_size, void* d_ws,
                              size_t ws_size, hipStream_t stream) {
  (void)in_sizes; (void)n_in; (void)out_size; (void)ws_size;
  const float* x = (const float*)d_in[0];
  float* out = (float*)d_out;
  int* fpsidx = (int*)d_ws;              // <= 8*1024 ints
  int* ballidx = fpsidx + 8 * 1024;      // <= 8*1024*32 ints

  // d_out flat layout: coords tuple then data tuple
  const long long o_c1 = 393216, o_c2 = 417792, o_c3 = 423936, o_c4 = 425472;
  const long long o_d0 = 425856, o_d1 = 1605504, o_d2 = 2129792,
                  o_d3 = 2391936, o_d4 = 2523008;
  const int B = 8;

  copy_strided3_kernel<<<(B * 16384 * 3 + 255) / 256, 256, 0, stream>>>(
      x, out, B * 16384);
  copy_flat_kernel<<<(int)((B * 16384LL * 9 + 255) / 256), 256, 0, stream>>>(
      x, out + o_d0, B * 16384LL * 9);

  // ---- SA layer 1: N=16384 (stride 9), S=1024, r=0.1, C: 12->32->32->64 ---
  {
    const int N = 16384, S = 1024;
    float* nxz = out + o_c1;
    fps_kernel<<<B, 256, (size_t)N * sizeof(float), stream>>>(x, 9, N, S,
                                                              fpsidx, nxz);
    ballquery_kernel<<<(B * S + 7) / 8, 256, 0, stream>>>(
        x, 9, N, S, nxz, 0.01f, ballidx, B * S);
    sa_mlp_kernel<12, 12, 32, 32, 64><<<B * S, 256, 0, stream>>>(
        x, 9, x, 9, nxz, ballidx, (const float*)d_in[1], (const float*)d_in[2],
        (const float*)d_in[3], (const float*)d_in[4], (const float*)d_in[5],
        (const float*)d_in[6], out + o_d1, N, S);
  }
  // ---- SA layer 2: N=1024, S=256, r=0.2, C: 67->64->64->128 ---------------
  {
    const int N = 1024, S = 256;
    const float* coords = out + o_c1;
    const float* feats = out + o_d1;
    float* nxz = out + o_c2;
    fps_kernel<<<B, 256, (size_t)N * sizeof(float), stream>>>(coords, 3, N, S,
                                                              fpsidx, nxz);
    ballquery_kernel<<<(B * S + 7) / 8, 256, 0, stream>>>(
        coords, 3, N, S, nxz, 0.04f, ballidx, B * S);
    sa_mlp_kernel<67, 68, 64, 64, 128><<<B * S, 256, 0, stream>>>(
        coords, 3, feats, 64, nxz, ballidx, (const float*)d_in[7],
        (const float*)d_in[8], (const float*)d_in[9], (const float*)d_in[10],
        (const float*)d_in[11], (const float*)d_in[12], out + o_d2, N, S);
  }
  // ---- SA layer 3: N=256, S=64, r=0.4, C: 131->128->128->256 --------------
  {
    const int N = 256, S = 64;
    const float* coords = out + o_c2;
    const float* feats = out + o_d2;
    float* nxz = out + o_c3;
    fps_kernel<<<B, 256, (size_t)N * sizeof(float), stream>>>(coords, 3, N, S,
                                                              fpsidx, nxz);
    ballquery_kernel<<<(B * S + 7) / 8, 256, 0, stream>>>(
        coords, 3, N, S, nxz, 0.16f, ballidx, B * S);
    sa_mlp_kernel<131, 132, 128, 128, 256><<<B * S, 256, 0, stream>>>(
        coords, 3, feats, 128, nxz, ballidx, (const float*)d_in[13],
        (const float*)d_in[14], (const float*)d_in[15], (const float*)d_in[16],
        (const float*)d_in[17], (const float*)d_in[18], out + o_d3, N, S);
  }
  // ---- SA layer 4: N=64, S=16, r=0.8, C: 259->256->256->512 ---------------
  {
    const int N = 64, S = 16;
    const float* coords = out + o_c3;
    const float* feats = out + o_d3;
    float* nxz = out + o_c4;
    fps_kernel<<<B, 256, (size_t)N * sizeof(float), stream>>>(coords, 3, N, S,
                                                              fpsidx, nxz);
    ballquery_kernel<<<(B * S + 7) / 8, 256, 0, stream>>>(
        coords, 3, N, S, nxz, 0.64f, ballidx, B * S);
    sa_mlp_kernel<259, 260, 256, 256, 512><<<B * S, 256, 0, stream>>>(
        coords, 3, feats, 256, nxz, ballidx, (const float*)d_in[19],
        (const float*)d_in[20], (const float*)d_in[21], (const float*)d_in[22],
        (const float*)d_in[23], (const float*)d_in[24], out + o_d4, N, S);
  }
}